// Interaction_17600775979372
// MI455X (gfx1250) — compile-verified
//
#include <hip/hip_runtime.h>
#include <math.h>

#define NN 16384
#define EE 262144
#define CC 32
#define DD 9
#define BB 8
#define HH 64
#define ZZ 4
#define HSTR 65   // padded LDS row stride (64 banks -> avoid stride-64 conflicts)

typedef float v2f __attribute__((ext_vector_type(2)));
typedef float v8f __attribute__((ext_vector_type(8)));

// D(16x16,f32) = A(16x4,f32) * B(4x16,f32) + C   — CDNA5 fp32 WMMA
__device__ __forceinline__ v8f wmma4(v2f a, v2f b, v8f c) {
  return __builtin_amdgcn_wmma_f32_16x16x4_f32(
      /*neg_a=*/false, a, /*neg_b=*/false, b,
      /*c_mod=*/(short)0, c, /*reuse_a=*/false, /*reuse_b=*/false);
}

// fast silu: x * v_rcp_f32(1+exp(-x))  (avoid IEEE div expansion)
__device__ __forceinline__ float silu(float x) {
  return x * __builtin_amdgcn_rcpf(1.0f + __expf(-x));
}

// ---------------------------------------------------------------------------
// Kernel 1: x = block_linear(node_feats, W_up)   [N,C,D]
// block = 1 node, 288 threads (q,d); W_up staged in LDS
// ---------------------------------------------------------------------------
__global__ void k_linear_up(const float* __restrict__ nf,
                            const float* __restrict__ Wup,
                            float* __restrict__ xo) {
  __shared__ float s_in[CC * DD];
  __shared__ float s_w[3 * CC * CC];
  const int n = blockIdx.x;
  const int t = threadIdx.x;          // 0..287
  s_in[t] = nf[(size_t)n * CC * DD + t];
  for (int i = t; i < 3 * CC * CC; i += CC * DD) s_w[i] = Wup[i];
  __syncthreads();
  const int q = t / DD, d = t % DD;
  const int l = (d == 0) ? 0 : (d < 4 ? 1 : 2);
  float acc = 0.f;
#pragma unroll
  for (int c = 0; c < CC; ++c) acc += s_in[c * DD + d] * s_w[l * CC * CC + c * CC + q];
  xo[(size_t)n * CC * DD + q * DD + d] = acc;
}

// ---------------------------------------------------------------------------
// Kernel 2: radial MLP with fp32 WMMA.  w = (silu(silu(ef@W1)@W2)@W3)*cutoff
// 128 threads = 4 waves; each wave owns a 16-edge tile. 104 WMMAs per wave.
// ---------------------------------------------------------------------------
__global__ void k_edge_mlp(const float* __restrict__ ef,
                           const float* __restrict__ W1,
                           const float* __restrict__ W2,
                           const float* __restrict__ W3,
                           const float* __restrict__ cutoff,
                           float* __restrict__ wout) {
  __shared__ float sW1[BB * HH];
  __shared__ float sW2[HH * HH];
  __shared__ float sW3[HH * CC];
  __shared__ float h1[4][16 * HSTR];
  __shared__ float h2[4][16 * HSTR];

  const int t = threadIdx.x;          // 0..127
  for (int i = t; i < BB * HH; i += 128) sW1[i] = W1[i];
  for (int i = t; i < HH * HH; i += 128) sW2[i] = W2[i];
  for (int i = t; i < HH * CC; i += 128) sW3[i] = W3[i];
  __syncthreads();

  const int wv = t >> 5, lane = t & 31;
  const int m = lane & 15;            // A row / C column index
  const int hi = lane >> 4;           // lane half
  const int koff = hi * 2;            // A/B K sub-offset for this lane half
  const int e0 = blockIdx.x * 64 + wv * 16;

  // ---- layer 1: [16x8] @ [8x64] -> silu -> h1 ----
  for (int nt = 0; nt < 4; ++nt) {
    v8f acc = {0, 0, 0, 0, 0, 0, 0, 0};
#pragma unroll
    for (int kk = 0; kk < 2; ++kk) {
      const int k = kk * 4 + koff;
      v2f a, b;
      a.x = ef[(size_t)(e0 + m) * BB + k];
      a.y = ef[(size_t)(e0 + m) * BB + k + 1];
      b.x = sW1[k * HH + nt * 16 + m];
      b.y = sW1[(k + 1) * HH + nt * 16 + m];
      acc = wmma4(a, b, acc);
    }
#pragma unroll
    for (int r = 0; r < 8; ++r) {
      const int M = r + hi * 8;
      h1[wv][M * HSTR + nt * 16 + m] = silu(acc[r]);
    }
  }
  __syncthreads();

  // ---- layer 2: [16x64] @ [64x64] -> silu -> h2 ----
  for (int nt = 0; nt < 4; ++nt) {
    v8f acc = {0, 0, 0, 0, 0, 0, 0, 0};
#pragma unroll
    for (int kk = 0; kk < 16; ++kk) {
      const int k = kk * 4 + koff;
      v2f a, b;
      a.x = h1[wv][m * HSTR + k];
      a.y = h1[wv][m * HSTR + k + 1];
      b.x = sW2[k * HH + nt * 16 + m];
      b.y = sW2[(k + 1) * HH + nt * 16 + m];
      acc = wmma4(a, b, acc);
    }
#pragma unroll
    for (int r = 0; r < 8; ++r) {
      const int M = r + hi * 8;
      h2[wv][M * HSTR + nt * 16 + m] = silu(acc[r]);
    }
  }
  __syncthreads();

  // ---- layer 3: [16x64] @ [64x32] * cutoff -> wout ----
  for (int nt = 0; nt < 2; ++nt) {
    v8f acc = {0, 0, 0, 0, 0, 0, 0, 0};
#pragma unroll
    for (int kk = 0; kk < 16; ++kk) {
      const int k = kk * 4 + koff;
      v2f a, b;
      a.x = h2[wv][m * HSTR + k];
      a.y = h2[wv][m * HSTR + k + 1];
      b.x = sW3[k * CC + nt * 16 + m];
      b.y = sW3[(k + 1) * CC + nt * 16 + m];
      acc = wmma4(a, b, acc);
    }
#pragma unroll
    for (int r = 0; r < 8; ++r) {
      const int M = r + hi * 8;
      const int e = e0 + M;
      wout[(size_t)e * CC + nt * 16 + m] = acc[r] * cutoff[e];
    }
  }
}

// ---------------------------------------------------------------------------
// Kernel 3: per-edge tensor product (WMMA) + atomic scatter into m_agg.
// 256 threads = 8 waves, 1 edge per wave.
//   per edge:  D[c,k] = (w[e,c] * xs[c,i]) @ M[i,k],  M = sum_j sh[j]*CG[:,j,:]
//   32x9 @ 9x9 -> 2 M-tiles x 3 K-steps of V_WMMA_F32_16X16X4_F32 (zero-padded)
// ---------------------------------------------------------------------------
__global__ void k_edge_tp(const float* __restrict__ x,
                          const float* __restrict__ sh,
                          const float* __restrict__ wE,
                          const float* __restrict__ CG,
                          const int* __restrict__ esrc,
                          const int* __restrict__ edst,
                          float* __restrict__ magg) {
  __shared__ float s_cg[DD * DD * DD];
  __shared__ float s_M[8][DD * DD];
  const int t = threadIdx.x;          // 0..255
  for (int i = t; i < DD * DD * DD; i += 256) s_cg[i] = CG[i];
  __syncthreads();

  const int wv = t >> 5, lane = t & 31;
  const int e = blockIdx.x * 8 + wv;
  const int m = lane & 15;
  const int hi = lane >> 4;
  const int koff = hi * 2;

  // lanes 0..8 build M[i,k] = sum_j sh[j] * CG[i,j,k] for this edge
  if (lane < DD) {
    const int i = lane;
#pragma unroll
    for (int k = 0; k < DD; ++k) {
      float acc = 0.f;
#pragma unroll
      for (int j = 0; j < DD; ++j)
        acc += sh[(size_t)e * DD + j] * s_cg[i * (DD * DD) + j * DD + k];
      s_M[wv][i * DD + k] = acc;
    }
  }
  __syncthreads();

  const int src = esrc[e];
  const int dst = edst[e];
  const float* xb = x + (size_t)src * CC * DD;
  const float wc0 = wE[(size_t)e * CC + m];        // row scale, channels 0..15
  const float wc1 = wE[(size_t)e * CC + 16 + m];   // row scale, channels 16..31

  v8f acc0 = {0, 0, 0, 0, 0, 0, 0, 0};
  v8f acc1 = {0, 0, 0, 0, 0, 0, 0, 0};
#pragma unroll
  for (int kb = 0; kb < 3; ++kb) {
    const int i0 = kb * 4 + koff, i1 = i0 + 1;
    v2f b, a0, a1;
    b.x  = (i0 < DD && m < DD) ? s_M[wv][i0 * DD + m] : 0.f;
    b.y  = (i1 < DD && m < DD) ? s_M[wv][i1 * DD + m] : 0.f;
    a0.x = (i0 < DD) ? xb[m * DD + i0] * wc0 : 0.f;
    a0.y = (i1 < DD) ? xb[m * DD + i1] * wc0 : 0.f;
    a1.x = (i0 < DD) ? xb[(16 + m) * DD + i0] * wc1 : 0.f;
    a1.y = (i1 < DD) ? xb[(16 + m) * DD + i1] * wc1 : 0.f;
    acc0 = wmma4(a0, b, acc0);
    acc1 = wmma4(a1, b, acc1);
  }

  // scatter: D rows = channels, cols = k (valid k < 9)
  if (m < DD) {
#pragma unroll
    for (int r = 0; r < 8; ++r) {
      const int ch = r + hi * 8;
      unsafeAtomicAdd(&magg[(size_t)dst * CC * DD + ch * DD + m], acc0[r]);
      unsafeAtomicAdd(&magg[(size_t)dst * CC * DD + (16 + ch) * DD + m], acc1[r]);
    }
  }
}

// ---------------------------------------------------------------------------
// Kernel 4: m_i = block_linear(m_agg, W_lin)/16 ; W_node = attrs@W_sc ;
//           scs = per-block (m_i @ W_node).   block = 1 node, 288 threads.
// ---------------------------------------------------------------------------
__global__ void k_node_out(const float* __restrict__ magg,
                           const float* __restrict__ attrs,
                           const float* __restrict__ Wlin,
                           const float* __restrict__ Wsc,
                           float* __restrict__ out_mi,
                           float* __restrict__ out_scs) {
  __shared__ float s_m[CC * DD];
  __shared__ float s_mi[CC * DD];
  __shared__ float s_wn[3 * CC * CC];
  __shared__ float s_wl[3 * CC * CC];
  const int n = blockIdx.x, t = threadIdx.x;   // 0..287
  s_m[t] = magg[(size_t)n * CC * DD + t];
  float az[ZZ];
#pragma unroll
  for (int z = 0; z < ZZ; ++z) az[z] = attrs[(size_t)n * ZZ + z];
  for (int idx = t; idx < 3 * CC * CC; idx += CC * DD) {
    float acc = 0.f;
#pragma unroll
    for (int z = 0; z < ZZ; ++z) acc += az[z] * Wsc[(size_t)z * 3 * CC * CC + idx];
    s_wn[idx] = acc;
    s_wl[idx] = Wlin[idx];
  }
  __syncthreads();

  const int q = t / DD, d = t % DD;
  const int l = (d == 0) ? 0 : (d < 4 ? 1 : 2);
  float acc = 0.f;
#pragma unroll
  for (int c = 0; c < CC; ++c) acc += s_m[c * DD + d] * s_wl[l * CC * CC + c * CC + q];
  acc *= (1.0f / 16.0f);               // avg_num_neighbors
  out_mi[(size_t)n * CC * DD + q * DD + d] = acc;
  s_mi[q * DD + d] = acc;              // store as channel=q for next stage
  __syncthreads();

  float acc2 = 0.f;
#pragma unroll
  for (int c = 0; c < CC; ++c) acc2 += s_mi[c * DD + d] * s_wn[l * CC * CC + c * CC + q];
  out_scs[(size_t)n * CC * DD + q * DD + d] = acc2;
}

// ---------------------------------------------------------------------------
extern "C" void kernel_launch(void* const* d_in, const int* in_sizes, int n_in,
                              void* d_out, int out_size, void* d_ws, size_t ws_size,
                              hipStream_t stream) {
  const float* node_feats = (const float*)d_in[0];
  const float* node_attrs = (const float*)d_in[1];
  const float* edge_feats = (const float*)d_in[2];
  const float* edge_sh    = (const float*)d_in[3];
  const float* cutoff     = (const float*)d_in[4];
  const float* W_up       = (const float*)d_in[5];
  const float* W1         = (const float*)d_in[6];
  const float* W2         = (const float*)d_in[7];
  const float* W3         = (const float*)d_in[8];
  const float* W_lin      = (const float*)d_in[9];
  const float* W_sc       = (const float*)d_in[10];
  const float* CG         = (const float*)d_in[11];
  const int*   esrc       = (const int*)d_in[12];
  const int*   edst       = (const int*)d_in[13];

  float* ws   = (float*)d_ws;
  float* x    = ws;                               // N*C*D
  float* wE   = x + (size_t)NN * CC * DD;         // E*C
  float* magg = wE + (size_t)EE * CC;             // N*C*D

  k_linear_up<<<NN, CC * DD, 0, stream>>>(node_feats, W_up, x);
  k_edge_mlp<<<EE / 64, 128, 0, stream>>>(edge_feats, W1, W2, W3, cutoff, wE);
  hipMemsetAsync(magg, 0, (size_t)NN * CC * DD * sizeof(float), stream);
  k_edge_tp<<<EE / 8, 256, 0, stream>>>(x, edge_sh, wE, CG, esrc, edst, magg);

  float* out_mi  = (float*)d_out;
  float* out_scs = out_mi + (size_t)NN * CC * DD;
  k_node_out<<<NN, CC * DD, 0, stream>>>(magg, node_attrs, W_lin, W_sc, out_mi, out_scs);
}